// MetricLoss_23682449670377
// MI455X (gfx1250) — compile-verified
//
#include <hip/hip_runtime.h>

typedef __attribute__((ext_vector_type(2))) float v2f;
typedef __attribute__((ext_vector_type(4))) float v4f;
typedef __attribute__((ext_vector_type(8))) float v8f;

static constexpr int D  = 128;    // feature dim (reference)
static constexpr int KG = 4;      // group size (BATCH_K)

// ---------------------------------------------------------------------------
// Kernel 1: per-row sum of squares; also zero the two double accumulators.
// ---------------------------------------------------------------------------
__global__ void sq_kernel(const float* __restrict__ x,
                          float* __restrict__ sq,
                          double* __restrict__ accum, int B) {
    int row = blockIdx.x * blockDim.x + threadIdx.x;
    if (row == 0) { accum[0] = 0.0; accum[1] = 0.0; }
    if (row < B) {
        const float4* p = (const float4*)(x + (size_t)row * D);
        float s = 0.f;
#pragma unroll
        for (int i = 0; i < D / 4; ++i) {
            float4 v = p[i];
            s += v.x * v.x + v.y * v.y + v.z * v.z + v.w * v.w;
        }
        sq[row] = s;
    }
}

__device__ __forceinline__ v2f lo2(v4f q) { v2f r; r[0] = q[0]; r[1] = q[1]; return r; }
__device__ __forceinline__ v2f hi2(v4f q) { v2f r; r[0] = q[2]; r[1] = q[3]; return r; }

// ---------------------------------------------------------------------------
// Map C/D VGPR layout -> (i,j), apply group masks + hinge.
// C layout (f32 16x16): vgpr r, lanes 0-15 -> M=r,N=lane; lanes 16-31 -> M=r+8.
// ---------------------------------------------------------------------------
__device__ __forceinline__ void process_tile(const v8f& acc, int ibase,
                                             const float* sqi, float sqj,
                                             int jcol, float& s_hom,
                                             float& s_het) {
    const int gj = jcol >> 2;
#pragma unroll
    for (int r = 0; r < 8; ++r) {
        int i = ibase + r;
        float d2 = sqi[r] + sqj - 2.0f * acc[r];
        int gi = i >> 2;
        if (gi < gj) s_het += fmaxf(1.0f - d2, 0.0f);
        if (gi == gj && i != jcol) s_hom += d2;
    }
}

// ---------------------------------------------------------------------------
// Kernel 2: one wave per 64x32 macro-tile (4x2 of 16x16 WMMA tiles).
// Per K-octet: 6x global_load_b128 feed 16x V_WMMA_F32_16X16X4_F32.
// K-permutation trick: lanes 0-15 carry K=kk..kk+3, lanes 16-31 K=kk+4..kk+7;
// chunk-even uses .xy halves, chunk-odd uses .zw halves (valid since the Gram
// sum is K-order-invariant and A/B share the same per-lane layout).
// ---------------------------------------------------------------------------
__global__ __launch_bounds__(128)
void gram_tile_kernel(const float* __restrict__ x,
                      const float* __restrict__ sq,
                      double* __restrict__ accum, int B) {
    const int lane = threadIdx.x & 31;
    const int wave = threadIdx.x >> 5;
    const int ntc  = B >> 5;                  // 32-wide col tiles
    const int ntr  = B >> 6;                  // 64-tall row tiles
    const int C = blockIdx.x * 4 + wave;
    const int R = blockIdx.y;
    if (R >= ntr || C >= ntc || C < 2 * R) return;   // wave-uniform: EXEC full

    const int i0 = R * 64;
    const int j0 = C * 32;

    const int mlane = lane & 15;
    const int koff  = (lane >> 4) * 4;        // lanes<16: K 0..3 ; lanes>=16: K 4..7

    const float* pa0 = x + (size_t)(i0 +  0 + mlane) * D + koff;
    const float* pa1 = x + (size_t)(i0 + 16 + mlane) * D + koff;
    const float* pa2 = x + (size_t)(i0 + 32 + mlane) * D + koff;
    const float* pa3 = x + (size_t)(i0 + 48 + mlane) * D + koff;
    const float* pb0 = x + (size_t)(j0 +  0 + mlane) * D + koff;
    const float* pb1 = x + (size_t)(j0 + 16 + mlane) * D + koff;

    v8f acc[4][2];
#pragma unroll
    for (int ti = 0; ti < 4; ++ti) {
        acc[ti][0] = (v8f){};
        acc[ti][1] = (v8f){};
    }

#pragma unroll 2
    for (int kk = 0; kk < D; kk += 8) {
        v4f qa0 = *(const v4f*)(pa0 + kk);
        v4f qa1 = *(const v4f*)(pa1 + kk);
        v4f qa2 = *(const v4f*)(pa2 + kk);
        v4f qa3 = *(const v4f*)(pa3 + kk);
        v4f qb0 = *(const v4f*)(pb0 + kk);
        v4f qb1 = *(const v4f*)(pb1 + kk);

        v2f al[4] = { lo2(qa0), lo2(qa1), lo2(qa2), lo2(qa3) };
        v2f bl[2] = { lo2(qb0), lo2(qb1) };
        v2f ah[4] = { hi2(qa0), hi2(qa1), hi2(qa2), hi2(qa3) };
        v2f bh[2] = { hi2(qb0), hi2(qb1) };

#pragma unroll
        for (int ti = 0; ti < 4; ++ti)
#pragma unroll
            for (int tj = 0; tj < 2; ++tj)
                acc[ti][tj] = __builtin_amdgcn_wmma_f32_16x16x4_f32(
                    false, al[ti], false, bl[tj], (short)0, acc[ti][tj], false, false);
#pragma unroll
        for (int ti = 0; ti < 4; ++ti)
#pragma unroll
            for (int tj = 0; tj < 2; ++tj)
                acc[ti][tj] = __builtin_amdgcn_wmma_f32_16x16x4_f32(
                    false, ah[ti], false, bh[tj], (short)0, acc[ti][tj], false, false);
    }

    // ---- masks + hinge + partial sums ----
    const int n_lo  = lane & 15;
    const int mbase = (lane >> 4) * 8;

    const int   jc[2]  = { j0 + n_lo, j0 + 16 + n_lo };
    const float sqj[2] = { sq[jc[0]], sq[jc[1]] };

    float s_hom = 0.f, s_het = 0.f;
#pragma unroll
    for (int ti = 0; ti < 4; ++ti) {
        const int ibase = i0 + ti * 16 + mbase;
        float sqi[8];
#pragma unroll
        for (int r = 0; r < 8; ++r) sqi[r] = sq[ibase + r];
#pragma unroll
        for (int tj = 0; tj < 2; ++tj)
            process_tile(acc[ti][tj], ibase, sqi, sqj[tj], jc[tj], s_hom, s_het);
    }

    // wave32 reduction
#pragma unroll
    for (int off = 16; off > 0; off >>= 1) {
        s_hom += __shfl_xor(s_hom, off, 32);
        s_het += __shfl_xor(s_het, off, 32);
    }
    if (lane == 0) {
        atomicAdd(&accum[0], (double)s_hom);
        atomicAdd(&accum[1], (double)s_het);
    }
}

// ---------------------------------------------------------------------------
// Kernel 3: divide by exact counts, emit the two losses.
// ---------------------------------------------------------------------------
__global__ void finalize_kernel(const double* __restrict__ accum,
                                float* __restrict__ out, int B) {
    if (blockIdx.x == 0 && threadIdx.x == 0) {
        double cnt_homo = (double)B * (double)(KG - 1);          // (B/k)*k*(k-1)
        double ngrp = (double)(B / KG);
        double cnt_het = ngrp * (ngrp - 1.0) * 0.5 * (double)(KG * KG);
        out[0] = (float)(accum[0] / cnt_homo);
        out[1] = (float)(accum[1] / cnt_het);
    }
}

// ---------------------------------------------------------------------------
extern "C" void kernel_launch(void* const* d_in, const int* in_sizes, int n_in,
                              void* d_out, int out_size, void* d_ws, size_t ws_size,
                              hipStream_t stream) {
    const float* x = (const float*)d_in[0];
    const int B = in_sizes[0] / D;                 // 8192

    double* accum = (double*)d_ws;                 // 2 doubles
    float*  sq    = (float*)((char*)d_ws + 256);   // B floats
    float*  out   = (float*)d_out;

    sq_kernel<<<(B + 255) / 256, 256, 0, stream>>>(x, sq, accum, B);

    const int ntc = B / 32;                        // 256 col tiles
    const int ntr = B / 64;                        // 128 row tiles
    dim3 grid(ntc / 4, ntr);                       // 4 waves (cols) per 128-thread block
    gram_tile_kernel<<<grid, 128, 0, stream>>>(x, sq, accum, B);

    finalize_kernel<<<1, 32, 0, stream>>>(accum, out, B);
}